// Confidence_82094004896480
// MI455X (gfx1250) — compile-verified
//
#include <hip/hip_runtime.h>

// Problem constants (match reference setup_inputs)
#define N_SEG   256
#define NCH     32
#define HW      (512 * 512)
#define N_IMG   8
#define ACC_ELEMS (2 * N_SEG * NCH)   // 16384 floats: [sum 8192][cnt 8192]

typedef __attribute__((ext_vector_type(2))) float v2f;
typedef __attribute__((ext_vector_type(8))) float v8f;

// ---------------------------------------------------------------------------
// Stage 1: per-workgroup private segment sums/counts in LDS, streamed once.
// Grid: N_IMG * G blocks, 256 threads (8 waves). Wave = 1 pixel, lane = channel.
// Image/slic are read exactly once (264 MB > useful L2 share) -> NT loads.
// LDS atomics: addresses s*32 + lane are 32 consecutive words -> 32 distinct
// banks, conflict-free within the wave.
// ---------------------------------------------------------------------------
__global__ __launch_bounds__(256) void seg_partial_kernel(
    const float* __restrict__ image, const int* __restrict__ slic,
    float* __restrict__ partials, int G)
{
    __shared__ float acc[ACC_ELEMS];   // 64 KB

    const int tid  = threadIdx.x;
    const int img  = blockIdx.x / G;
    const int wg   = blockIdx.x % G;

    for (int i = tid; i < ACC_ELEMS; i += 256) acc[i] = 0.0f;
    __syncthreads();

    const int lane  = tid & 31;
    const int wave  = tid >> 5;
    const int chunk = HW / G;
    const int p0    = wg * chunk;
    const size_t img_pix = (size_t)img * HW;

    #pragma unroll 4
    for (int p = p0 + wave; p < p0 + chunk; p += 8) {
        const size_t pix = img_pix + (size_t)p;
        // Prefetch ~32 pixels ahead (gfx1250 global_prefetch_b8; OOB dropped)
        __builtin_prefetch(&image[(pix + 32) * NCH + lane], 0, 0);
        const float v = __builtin_nontemporal_load(&image[pix * NCH + lane]); // 128B/wave, TH=NT
        const int   s = __builtin_nontemporal_load(&slic[pix]) - 1;           // broadcast load
        atomicAdd(&acc[s * NCH + lane], v);                                   // ds_add_f32
        atomicAdd(&acc[N_SEG * NCH + s * NCH + lane],
                  (v != 0.0f) ? 1.0f : 0.0f);                                 // ds_add_f32
    }
    __syncthreads();

    float* dst = partials + (size_t)(img * G + wg) * ACC_ELEMS;
    for (int i = tid; i < ACC_ELEMS; i += 256) dst[i] = acc[i];
}

// ---------------------------------------------------------------------------
// Stage 2: reduce G partials per image with V_WMMA_F32_16X16X4_F32 and divide.
// D = A(16x4, all ones) x B(4x16 partial rows) + C  => every row of D is the
// column sum; all-ones A makes the result invariant to B's K-slot layout
// (exact f32 arithmetic). One wave handles one 16-column tile (sum + cnt).
// ---------------------------------------------------------------------------
__global__ __launch_bounds__(256) void seg_reduce_wmma_kernel(
    const float* __restrict__ partials, float* __restrict__ out, int G)
{
    const int tid  = threadIdx.x;
    const int lane = tid & 31;
    const int w    = blockIdx.x * 8 + (tid >> 5);  // global wave id
    const int img  = w >> 9;                       // 512 tiles per image
    const int tile = w & 511;
    const int col  = tile * 16 + (lane & 15);
    const int half = lane >> 4;

    v2f a; a.x = 1.0f; a.y = 1.0f;                 // all-ones A (16x4)
    v8f cs = {};                                   // sum accumulator (16x16 f32)
    v8f cc = {};                                   // cnt accumulator

    const float* base = partials + (size_t)img * G * ACC_ELEMS;
    const int ksteps = G >> 2;                     // K = G, 4 per WMMA
    for (int kk = 0; kk < ksteps; ++kk) {
        const int r0 = kk * 4 + half;
        const int r1 = kk * 4 + 2 + half;
        v2f bs, bc;
        bs.x = __builtin_nontemporal_load(&base[(size_t)r0 * ACC_ELEMS + col]);
        bs.y = __builtin_nontemporal_load(&base[(size_t)r1 * ACC_ELEMS + col]);
        bc.x = __builtin_nontemporal_load(&base[(size_t)r0 * ACC_ELEMS + N_SEG * NCH + col]);
        bc.y = __builtin_nontemporal_load(&base[(size_t)r1 * ACC_ELEMS + N_SEG * NCH + col]);
        cs = __builtin_amdgcn_wmma_f32_16x16x4_f32(false, a, false, bs,
                                                   (short)0, cs, false, false);
        cc = __builtin_amdgcn_wmma_f32_16x16x4_f32(false, a, false, bc,
                                                   (short)0, cc, false, false);
    }

    // Every lane's c[0] holds the column sum for N = lane&15 (rows identical).
    if (lane < 16) {
        out[(size_t)img * (N_SEG * NCH) + col] = cs[0] / cc[0];
    }
}

// ---------------------------------------------------------------------------
extern "C" void kernel_launch(void* const* d_in, const int* in_sizes, int n_in,
                              void* d_out, int out_size, void* d_ws, size_t ws_size,
                              hipStream_t stream) {
    const float* image = (const float*)d_in[0];
    const int*   slic  = (const int*)d_in[1];
    float*       out   = (float*)d_out;
    float*       ws    = (float*)d_ws;

    // Partials footprint: N_IMG * G * ACC_ELEMS * 4 bytes = G * 512 KiB.
    int G;
    if      (ws_size >= (size_t)64 * 524288) G = 64;   // 32 MB, 512 blocks
    else if (ws_size >= (size_t)32 * 524288) G = 32;   // 16 MB
    else if (ws_size >= (size_t)16 * 524288) G = 16;   //  8 MB
    else if (ws_size >= (size_t)8  * 524288) G = 8;    //  4 MB
    else                                     G = 4;    //  2 MB

    seg_partial_kernel<<<N_IMG * G, 256, 0, stream>>>(image, slic, ws, G);

    const int total_waves = N_IMG * (N_SEG * NCH / 16);  // 4096
    seg_reduce_wmma_kernel<<<total_waves / 8, 256, 0, stream>>>(ws, out, G);
}